// GAT_20899310863186
// MI455X (gfx1250) — compile-verified
//
#include <hip/hip_runtime.h>
#include <stdint.h>

typedef __attribute__((ext_vector_type(2))) float v2f;
typedef __attribute__((ext_vector_type(8))) float v8f;

#define N_NODES 50000

// ---------------------------------------------------------------- degree ----
__global__ void k_fill1(float* __restrict__ p, int n) {
    int i = blockIdx.x * blockDim.x + threadIdx.x;
    if (i < n) p[i] = 1.0f;
}

__global__ void k_deg_accum(const int* __restrict__ dst, float* __restrict__ deg, int E) {
    int e = blockIdx.x * blockDim.x + threadIdx.x;
    if (e < E) atomicAdd(&deg[dst[e]], 1.0f);
}

__global__ void k_rsqrt_inplace(float* __restrict__ p, int n) {
    int i = blockIdx.x * blockDim.x + threadIdx.x;
    if (i < n) p[i] = rsqrtf(p[i]);
}

// ------------------------------------------------------------- WMMA GEMM ----
// H[M x F] = (RELU ? max(IN,0) : IN)[M x K] @ W[K x F]
// One wave per 16-row M tile; the wave accumulates all F/16 N tiles so each
// A fragment (v_wmma_f32_16x16x4_f32 layout: lanes 0-15 = rows, vgpr0/1 hold
// K=kb,kb+1 with kb = 0 (lanes 0-15) or 2 (lanes 16-31)) is reused F/16 times.
template <int K, int F, bool RELU>
__global__ __launch_bounds__(128) void k_gemm_wmma(const float* __restrict__ in,
                                                   const float* __restrict__ W,
                                                   float* __restrict__ H,
                                                   int Mtiles) {
    const int lane = threadIdx.x & 31;
    const int wave = threadIdx.x >> 5;
    const int mt = blockIdx.x * 4 + wave;
    if (mt >= Mtiles) return;

    const int lm = lane & 15;          // row (A) / col (B,C,D) within tile
    const int kb = (lane >> 4) << 1;   // 0 for lanes 0-15, 2 for lanes 16-31
    const int m0 = mt * 16;

    constexpr int NT = F / 16;
    v8f acc[NT];
#pragma unroll
    for (int t = 0; t < NT; ++t) acc[t] = (v8f){0.f, 0.f, 0.f, 0.f, 0.f, 0.f, 0.f, 0.f};

    const float* __restrict__ arow = in + (size_t)(m0 + lm) * K;

    for (int k0 = 0; k0 < K; k0 += 4) {
        float a0 = arow[k0 + kb];
        float a1 = arow[k0 + kb + 1];
        if (RELU) { a0 = fmaxf(a0, 0.0f); a1 = fmaxf(a1, 0.0f); }
        v2f a; a.x = a0; a.y = a1;
#pragma unroll
        for (int t = 0; t < NT; ++t) {
            v2f b;
            b.x = W[(size_t)(k0 + kb) * F + t * 16 + lm];
            b.y = W[(size_t)(k0 + kb + 1) * F + t * 16 + lm];
            acc[t] = __builtin_amdgcn_wmma_f32_16x16x4_f32(
                false, a, false, b, (short)0, acc[t], false, false);
        }
    }

    // C/D layout: vgpr r holds row m0 + r (lanes 0-15) / m0 + 8 + r (lanes 16-31)
    const int rbase = (lane < 16) ? 0 : 8;
#pragma unroll
    for (int t = 0; t < NT; ++t) {
#pragma unroll
        for (int r = 0; r < 8; ++r) {
            H[(size_t)(m0 + rbase + r) * F + t * 16 + lm] = acc[t][r];
        }
    }
}

// ----------------------------------------------- self-loop + bias (init) ----
__global__ void k_selfterm(const float* __restrict__ h, const float* __restrict__ dis,
                           const float* __restrict__ bias, float* __restrict__ agg,
                           int F, int total) {
    int i = blockIdx.x * blockDim.x + threadIdx.x;
    if (i >= total) return;
    int node = i / F;
    int f = i - node * F;
    float d = dis[node];
    agg[i] = h[i] * d * d + bias[f];
}

// --------------------------------------------------------- edge scatter -----
// agg[dst] += dis[src]*dis[dst] * h[src]; F lanes cooperate on one edge.
template <int F>
__global__ __launch_bounds__(256) void k_scatter(const float* __restrict__ h,
                                                 const int* __restrict__ src,
                                                 const int* __restrict__ dst,
                                                 const float* __restrict__ dis,
                                                 float* __restrict__ agg, int E) {
    constexpr int EPB = 256 / F;
    int e = blockIdx.x * EPB + (threadIdx.x / F);
    int f = threadIdx.x & (F - 1);
    if (e >= E) return;
    int s = src[e];
    int d = dst[e];
    float c = dis[s] * dis[d];
    atomicAdd(&agg[(size_t)d * F + f], h[(size_t)s * F + f] * c);
}

// ---------------------------------------------------------------- launch ----
extern "C" void kernel_launch(void* const* d_in, const int* in_sizes, int n_in,
                              void* d_out, int out_size, void* d_ws, size_t ws_size,
                              hipStream_t stream) {
    const float* x  = (const float*)d_in[0];
    const int*   ei = (const int*)d_in[1];
    const float* W1 = (const float*)d_in[2];
    const float* b1 = (const float*)d_in[3];
    const float* W2 = (const float*)d_in[4];
    const float* b2 = (const float*)d_in[5];
    const float* W3 = (const float*)d_in[6];
    const float* b3 = (const float*)d_in[7];
    float* out = (float*)d_out;

    const int E = in_sizes[1] / 2;
    const int* src = ei;
    const int* dst = ei + E;

    char* ws = (char*)d_ws;
    float* dis  = (float*)ws;                                  // 50000 f32
    float* bufA = (float*)(ws + 200192);                       // 50000*128 f32 (h)
    float* bufB = (float*)(ws + 200192 + 25600000);            // 50000*128 f32 (agg1 / in2)
    float* bufC = (float*)(ws + 200192 + 51200000);            // 50000*64  f32 (agg2 / in3)

    const int Mtiles = N_NODES / 16;                           // 3125
    const int gemmBlocks = (Mtiles + 3) / 4;

    // deg -> dis = rsqrt(1 + indeg)
    k_fill1<<<(N_NODES + 255) / 256, 256, 0, stream>>>(dis, N_NODES);
    k_deg_accum<<<(E + 255) / 256, 256, 0, stream>>>(dst, dis, E);
    k_rsqrt_inplace<<<(N_NODES + 255) / 256, 256, 0, stream>>>(dis, N_NODES);

    // ---- layer 1: h1 = x @ W1 ; agg1 = scatter + self + b1
    k_gemm_wmma<256, 128, false><<<gemmBlocks, 128, 0, stream>>>(x, W1, bufA, Mtiles);
    k_selfterm<<<(N_NODES * 128 + 255) / 256, 256, 0, stream>>>(bufA, dis, b1, bufB, 128, N_NODES * 128);
    k_scatter<128><<<(E + 1) / 2, 256, 0, stream>>>(bufA, src, dst, dis, bufB, E);

    // ---- layer 2: h2 = relu(agg1) @ W2 ; agg2
    k_gemm_wmma<128, 64, true><<<gemmBlocks, 128, 0, stream>>>(bufB, W2, bufA, Mtiles);
    k_selfterm<<<(N_NODES * 64 + 255) / 256, 256, 0, stream>>>(bufA, dis, b2, bufC, 64, N_NODES * 64);
    k_scatter<64><<<(E + 3) / 4, 256, 0, stream>>>(bufA, src, dst, dis, bufC, E);

    // ---- layer 3: h3 = relu(agg2) @ W3 ; out
    k_gemm_wmma<64, 16, true><<<gemmBlocks, 128, 0, stream>>>(bufC, W3, bufA, Mtiles);
    k_selfterm<<<(N_NODES * 16 + 255) / 256, 256, 0, stream>>>(bufA, dis, b3, out, 16, N_NODES * 16);
    k_scatter<16><<<(E + 15) / 16, 256, 0, stream>>>(bufA, src, dst, dis, out, E);
}